// PCALayer_5583457485190
// MI455X (gfx1250) — compile-verified
//
#include <hip/hip_runtime.h>
#include <hip/hip_bf16.h>

typedef __attribute__((ext_vector_type(2))) float v2f;
typedef __attribute__((ext_vector_type(8))) float v8f;

static constexpr int BATCH = 16;
static constexpr int NPIX  = 16384;   // 128*128
static constexpr int CH    = 128;
static constexpr int NCOMP = 64;
static constexpr int SJ    = 129;     // LDS row stride (129 mod 64 == 1 -> conflict-free)

// ---------------------------------------------------------------------------
// Kernel 1: per-(batch,channel) mean over N.  One streaming pass of x.
// ---------------------------------------------------------------------------
__global__ void pca_mean_kernel(const float* __restrict__ x, float* __restrict__ mean) {
  const int b    = blockIdx.x;
  const int t    = threadIdx.x;        // 256 threads = 8 waves
  const int c    = t & 127;
  const int half = t >> 7;             // two n-phases per channel
  const float* xb = x + (size_t)b * NPIX * CH;
  float s = 0.0f;
  for (int n = half; n < NPIX; n += 2)
    s += xb[(size_t)n * CH + c];
  __shared__ float red[256];
  red[t] = s;
  __syncthreads();
  if (t < CH)
    mean[b * CH + c] = (red[c] + red[c + 128]) * (1.0f / (float)NPIX);
}

// ---------------------------------------------------------------------------
// Kernel 2: Gram matrix via V_WMMA_F32_16X16X4_F32, then mean correction:
//   cov = (G - N*mu*mu^T) / (N-1).
// grid = BATCH*64 (8x8 tiles of 16x16), 8 waves split K = NPIX.
// ---------------------------------------------------------------------------
__global__ void pca_cov_kernel(const float* __restrict__ x,
                               const float* __restrict__ mean,
                               float* __restrict__ cov) {
  const int blk  = blockIdx.x;
  const int b    = blk >> 6;
  const int tile = blk & 63;
  const int c0   = (tile >> 3) << 4;   // row channel strip
  const int c1   = (tile & 7) << 4;    // col channel strip
  const int wave = threadIdx.x >> 5;
  const int lane = threadIdx.x & 31;
  const float* xb = x + (size_t)b * NPIX * CH;

  // A matrix (16 x 4, M=channels, K=pixels): lane layout per ISA 7.12.2
  const int mA = lane & 15;
  const int kA = (lane >> 4) << 1;     // 0 or 2
  const int cA = c0 + mA;
  // B matrix (4 x 16, K=pixels, N=channels): row-striped across lanes
  const int nB = lane & 15;
  const int kB = lane >> 4;            // 0 or 1
  const int cB = c1 + nB;

  v8f acc = {};
  const int nBeg = wave * (NPIX / 8);
  const int nEnd = nBeg + (NPIX / 8);
  #pragma unroll 4
  for (int n = nBeg; n < nEnd; n += 4) {
    __builtin_prefetch(&xb[(size_t)(n + 32) * CH + cA], 0, 1);
    v2f a, bm;
    a.x  = xb[(size_t)(n + kA)     * CH + cA];
    a.y  = xb[(size_t)(n + kA + 1) * CH + cA];
    bm.x = xb[(size_t)(n + kB)     * CH + cB];
    bm.y = xb[(size_t)(n + kB + 2) * CH + cB];
    acc = __builtin_amdgcn_wmma_f32_16x16x4_f32(false, a, false, bm,
                                                (short)0, acc, false, false);
  }

  // cross-wave K reduction through LDS
  __shared__ float red[8 * 256];
  #pragma unroll
  for (int r = 0; r < 8; ++r) red[wave * 256 + lane * 8 + r] = acc[r];
  __syncthreads();

  const int e = threadIdx.x;           // 0..255, one C-tile element each
  float s = 0.0f;
  #pragma unroll
  for (int w = 0; w < 8; ++w) s += red[w * 256 + e];
  const int l   = e >> 3;
  const int r   = e & 7;
  const int row = c0 + r + ((l >> 4) << 3);   // C/D layout: M = r + 8*(lane>=16)
  const int col = c1 + (l & 15);
  const float mu_r = mean[b * CH + row];
  const float mu_c = mean[b * CH + col];
  cov[((size_t)b * CH + row) * CH + col] =
      (s - (float)NPIX * mu_r * mu_c) * (1.0f / (float)(NPIX - 1));
}

// ---------------------------------------------------------------------------
// Kernel 3: parallel cyclic Jacobi eigensolver, one workgroup (4 waves) per
// batch.  A and V live in LDS (2 * 128*129 floats = 129 KB of the 320 KB WGP
// LDS).  Tournament pairing gives 64 independent Givens rotations per round.
// Emits the top-64 eigenvector columns (descending eigenvalue) to wtop.
// ---------------------------------------------------------------------------
__global__ void pca_jacobi_kernel(const float* __restrict__ cov,
                                  float* __restrict__ wtop) {
  extern __shared__ float lds[];
  float* A = lds;                     // 128 x SJ
  float* V = lds + CH * SJ;           // 128 x SJ
  __shared__ int   pj[64], qj[64];
  __shared__ float cs[64], sn[64];
  __shared__ float evs[CH];

  const int b = blockIdx.x;
  const int t = threadIdx.x;          // 128 threads
  const float* cb = cov + (size_t)b * CH * CH;

  for (int i = t; i < CH * CH; i += 128) {
    const int r = i >> 7, c = i & 127;
    A[r * SJ + c] = cb[i];
    V[r * SJ + c] = (r == c) ? 1.0f : 0.0f;
  }
  __syncthreads();

  for (int sweep = 0; sweep < 8; ++sweep) {
    for (int round = 0; round < CH - 1; ++round) {
      if (t < 64) {                   // rotation setup: 64 disjoint pairs
        int p, q;
        if (t == 0) { p = CH - 1; q = round % (CH - 1); }
        else {
          p = (round + t) % (CH - 1);
          q = (round + (CH - 1) - t) % (CH - 1);
        }
        const float app = A[p * SJ + p];
        const float aqq = A[q * SJ + q];
        const float apq = A[p * SJ + q];
        float cc = 1.0f, ss = 0.0f;
        if (__builtin_fabsf(apq) > 1e-20f) {
          const float theta = 0.5f * (aqq - app) / apq;
          const float tt = (theta >= 0.0f ? 1.0f : -1.0f) /
                           (__builtin_fabsf(theta) + sqrtf(1.0f + theta * theta));
          cc = rsqrtf(1.0f + tt * tt);
          ss = tt * cc;
        }
        pj[t] = p; qj[t] = q; cs[t] = cc; sn[t] = ss;
      }
      __syncthreads();
      // row rotations: thread t owns column t (bank-conflict-free, stride SJ)
      for (int j = 0; j < 64; ++j) {
        const int p = pj[j], q = qj[j];
        const float cc = cs[j], ss = sn[j];
        const float ap = A[p * SJ + t], aq = A[q * SJ + t];
        A[p * SJ + t] = cc * ap - ss * aq;
        A[q * SJ + t] = ss * ap + cc * aq;
      }
      __syncthreads();
      // column rotations on A and V: thread t owns row t
      for (int j = 0; j < 64; ++j) {
        const int p = pj[j], q = qj[j];
        const float cc = cs[j], ss = sn[j];
        const float ap = A[t * SJ + p], aq = A[t * SJ + q];
        A[t * SJ + p] = cc * ap - ss * aq;
        A[t * SJ + q] = ss * ap + cc * aq;
        const float vp = V[t * SJ + p], vq = V[t * SJ + q];
        V[t * SJ + p] = cc * vp - ss * vq;
        V[t * SJ + q] = ss * vp + cc * vq;
      }
      __syncthreads();
    }
  }

  // rank eigenvalues (descending) and scatter top-64 eigenvector columns
  evs[t] = A[t * SJ + t];
  __syncthreads();
  const float mine = evs[t];
  int rank = 0;
  for (int i = 0; i < CH; ++i) {
    const float o = evs[i];
    rank += (o > mine) || (o == mine && i < t);
  }
  if (rank < NCOMP) {
    float* wb = wtop + (size_t)b * CH * NCOMP;
    for (int r = 0; r < CH; ++r)
      wb[r * NCOMP + rank] = V[r * SJ + t];
  }
}

// ---------------------------------------------------------------------------
// Kernel 4: projection (X - mu) @ Wtop via V_WMMA_F32_16X16X4_F32.
// grid = BATCH*1024 row tiles of 16 pixels; 4 waves cover the 4 16-col tiles.
// ---------------------------------------------------------------------------
__global__ void pca_proj_kernel(const float* __restrict__ x,
                                const float* __restrict__ mean,
                                const float* __restrict__ wtop,
                                float* __restrict__ out) {
  const int blk  = blockIdx.x;
  const int b    = blk >> 10;
  const int n0   = (blk & 1023) << 4;
  const int wave = threadIdx.x >> 5;   // 4 waves
  const int lane = threadIdx.x & 31;
  const int k0   = wave << 4;

  const float* xb = x + (size_t)b * NPIX * CH;
  const float* wb = wtop + (size_t)b * CH * NCOMP;
  const float* mu = mean + b * CH;

  const int mA    = lane & 15;         // pixel row within tile
  const int kAoff = (lane >> 4) << 1;  // 0 or 2
  const int nB    = lane & 15;
  const int kBoff = lane >> 4;         // 0 or 1

  v8f acc = {};
  #pragma unroll 4
  for (int c = 0; c < CH; c += 4) {
    v2f a, bm;
    const size_t rowBase = (size_t)(n0 + mA) * CH;
    a.x  = xb[rowBase + c + kAoff]     - mu[c + kAoff];
    a.y  = xb[rowBase + c + kAoff + 1] - mu[c + kAoff + 1];
    bm.x = wb[(size_t)(c + kBoff)     * NCOMP + k0 + nB];
    bm.y = wb[(size_t)(c + kBoff + 2) * NCOMP + k0 + nB];
    acc = __builtin_amdgcn_wmma_f32_16x16x4_f32(false, a, false, bm,
                                                (short)0, acc, false, false);
  }

  float* ob = out + ((size_t)b * NPIX + n0) * NCOMP;
  #pragma unroll
  for (int r = 0; r < 8; ++r) {
    const int row = r + ((lane >> 4) << 3);
    const int col = lane & 15;
    ob[(size_t)row * NCOMP + k0 + col] = acc[r];
  }
}

// ---------------------------------------------------------------------------
extern "C" void kernel_launch(void* const* d_in, const int* in_sizes, int n_in,
                              void* d_out, int out_size, void* d_ws, size_t ws_size,
                              hipStream_t stream) {
  (void)in_sizes; (void)n_in; (void)out_size; (void)ws_size;
  const float* x = (const float*)d_in[0];
  float* out = (float*)d_out;

  // workspace layout (fp32): mean | cov | wtop  (~1.6 MB total)
  float* mean = (float*)d_ws;
  float* cov  = mean + BATCH * CH;
  float* wtop = cov + (size_t)BATCH * CH * CH;

  pca_mean_kernel<<<dim3(BATCH), dim3(256), 0, stream>>>(x, mean);
  pca_cov_kernel<<<dim3(BATCH * 64), dim3(256), 0, stream>>>(x, mean, cov);
  pca_jacobi_kernel<<<dim3(BATCH), dim3(128),
                      (size_t)(2 * CH * SJ) * sizeof(float), stream>>>(cov, wtop);
  pca_proj_kernel<<<dim3(BATCH * 1024), dim3(128), 0, stream>>>(x, mean, wtop, out);
}